// CayleyStringPE_33380485824719
// MI455X (gfx1250) — compile-verified
//
#include <hip/hip_runtime.h>
#include <math.h>

#define DM 128
#define NPOS 8192
#define ROWS_PER_TENSOR (8 * 8192)        /* B*N = 65536 */
#define TOTAL_ROWS (2 * ROWS_PER_TENSOR)  /* q rows then k rows */
#define TILES (TOTAL_ROWS / 16)           /* 8192 16-row tiles */
#define AT_STRIDE 132                     /* conflict-free A-frag reads */
#define TAB_ENTRIES (NPOS * 64)           /* 524288 (cos,sin) pairs */
#define TAB_BLOCKS 512

typedef __attribute__((ext_vector_type(2))) float v2f;
typedef __attribute__((ext_vector_type(4))) float v4f;
typedef __attribute__((ext_vector_type(8))) float v8f;

// ---------------------------------------------------------------------------
// Prep kernel.
//   block 0        : W = (I - S)^(-1)(I + S) via rank-4 blocked Gauss-Jordan
//   blocks 1..512  : fill the (cos,sin) RoPE table (runs concurrently)
// ---------------------------------------------------------------------------
__global__ __launch_bounds__(256) void cayley_prep_kernel(
    const float* __restrict__ s_params, const float* __restrict__ pos,
    const float* __restrict__ freqs, float* __restrict__ W,
    float* __restrict__ tab) {
  const int tid = threadIdx.x;

  if (blockIdx.x != 0) {
    // ---- (cos,sin) table: tab[(n*64+p)*2 + {0,1}] = {cos, sin} ----------
    for (int e = (int)(blockIdx.x - 1) * 256 + tid; e < TAB_ENTRIES;
         e += TAB_BLOCKS * 256) {
      int n = e >> 6, p = e & 63;
      float ang = pos[n] * freqs[p];
      float sn, cs;
      __sincosf(ang, &sn, &cs);
      v2f sc;
      sc.x = cs;
      sc.y = sn;
      *(v2f*)(tab + 2 * e) = sc;
    }
    return;
  }

  // ---- block 0: blocked Gauss-Jordan ------------------------------------
  __shared__ __attribute__((aligned(16))) float Ash[DM * 130];
  __shared__ __attribute__((aligned(16))) float Bsh[DM * 130];
  __shared__ __attribute__((aligned(16))) float Lcol[DM * 4];  // A[:,pb..pb+3]
  __shared__ float Dinv[16];                                   // 4x4 inverse

  // Build A = I - S and B = I + S from the skew parameter vector.
  for (int f = tid; f < DM * DM; f += 256) {
    int i = f >> 7, j = f & 127;
    float s = 0.f;
    if (i < j) {
      int idx = i * (2 * DM - i - 1) / 2 + (j - i - 1);
      s = s_params[idx];
    } else if (i > j) {
      int idx = j * (2 * DM - j - 1) / 2 + (i - j - 1);
      s = -s_params[idx];
    }
    float diag = (i == j) ? 1.f : 0.f;
    Ash[i * 130 + j] = diag - s;
    Bsh[i * 130 + j] = diag + s;
  }
  __syncthreads();

  // Thread t owns column t of [A | B] (waves 0-3 -> A, 4-7 -> B).
  float* base = (tid < DM) ? (Ash + tid) : (Bsh + (tid - DM));

  for (int pb = 0; pb < DM; pb += 4) {
    // Snapshot multipliers L = A[:, pb..pb+3].
    for (int idx = tid; idx < DM * 4; idx += 256) {
      int i = idx >> 2, t = idx & 3;
      Lcol[i * 4 + t] = Ash[i * 130 + pb + t];
    }
    __syncthreads();

    // One lane inverts the 4x4 pivot block D = A[P,P] (no pivoting needed:
    // diag = 1 +- O(0.02) terms, S is a small skew perturbation).
    if (tid == 0) {
      float Dm[4][4], Di[4][4];
#pragma unroll
      for (int t = 0; t < 4; ++t)
#pragma unroll
        for (int u = 0; u < 4; ++u) {
          Dm[t][u] = Lcol[(pb + t) * 4 + u];
          Di[t][u] = (t == u) ? 1.f : 0.f;
        }
#pragma unroll
      for (int p = 0; p < 4; ++p) {
        float piv = 1.0f / Dm[p][p];
#pragma unroll
        for (int u = 0; u < 4; ++u) {
          Dm[p][u] *= piv;
          Di[p][u] *= piv;
        }
#pragma unroll
        for (int i2 = 0; i2 < 4; ++i2) {
          if (i2 == p) continue;
          float fq = Dm[i2][p];
#pragma unroll
          for (int u = 0; u < 4; ++u) {
            Dm[i2][u] = fmaf(-fq, Dm[p][u], Dm[i2][u]);
            Di[i2][u] = fmaf(-fq, Di[p][u], Di[i2][u]);
          }
        }
      }
#pragma unroll
      for (int t = 0; t < 4; ++t)
#pragma unroll
        for (int u = 0; u < 4; ++u) Dinv[4 * t + u] = Di[t][u];
    }
    __syncthreads();

    // New pivot rows for this column: q = Dinv * r.
    float r0 = base[(pb + 0) * 130];
    float r1 = base[(pb + 1) * 130];
    float r2 = base[(pb + 2) * 130];
    float r3 = base[(pb + 3) * 130];
    float q0 = Dinv[0] * r0 + Dinv[1] * r1 + Dinv[2] * r2 + Dinv[3] * r3;
    float q1 = Dinv[4] * r0 + Dinv[5] * r1 + Dinv[6] * r2 + Dinv[7] * r3;
    float q2 = Dinv[8] * r0 + Dinv[9] * r1 + Dinv[10] * r2 + Dinv[11] * r3;
    float q3 = Dinv[12] * r0 + Dinv[13] * r1 + Dinv[14] * r2 + Dinv[15] * r3;

    // Rank-4 trailing update: row_i -= L[i,:] . q  (rows outside pivot block)
    for (int i = 0; i < pb; ++i) {
      v4f l = *(const v4f*)(Lcol + i * 4);  // broadcast b128
      base[i * 130] -= l.x * q0 + l.y * q1 + l.z * q2 + l.w * q3;
    }
    for (int i = pb + 4; i < DM; ++i) {
      v4f l = *(const v4f*)(Lcol + i * 4);
      base[i * 130] -= l.x * q0 + l.y * q1 + l.z * q2 + l.w * q3;
    }
    base[(pb + 0) * 130] = q0;
    base[(pb + 1) * 130] = q1;
    base[(pb + 2) * 130] = q2;
    base[(pb + 3) * 130] = q3;
    __syncthreads();
  }

  for (int f = tid; f < DM * DM; f += 256)
    W[f] = Bsh[(f >> 7) * 130 + (f & 127)];
}

// ---------------------------------------------------------------------------
// Kernel 2: out = RoPE(h) @ W, fused; one 16x128 row tile per wave.
// Grid: 1024 blocks x 8 waves == 8192 tiles (q rows then k rows).
// ---------------------------------------------------------------------------
__global__ __launch_bounds__(256) void rope_cayley_gemm_kernel(
    const float* __restrict__ q, const float* __restrict__ k,
    const float* __restrict__ tab, const float* __restrict__ W,
    float* __restrict__ out) {
  // W pre-swizzled into per-lane B-fragment order: Wl[kb][nt][lane][2]
  __shared__ __attribute__((aligned(16))) float Wl[16384];
  // Per-wave RoPE'd 16x128 A tile, padded stride for bank-conflict-free frags
  __shared__ __attribute__((aligned(16))) float At[8][16 * AT_STRIDE];

  const int tid = threadIdx.x;
  const int lane = tid & 31;
  const int wave = tid >> 5;

  // ---- Stage W: coalesced b128 global reads, LDS scatter -----------------
  // Fragment semantics (ISA 16x4 layout mirrored for B=4x16):
  //   lane L holds (K = 4*kb + 2*(L>>4) + c, N = 16*nt + (L&15)), c = 0,1
  for (int f4 = tid; f4 < 4096; f4 += 256) {
    v4f w = *(const v4f*)(W + 4 * f4);
    int krow = f4 >> 5;
    int col0 = (f4 & 31) * 4;
    int kb = krow >> 2, c = krow & 1, hb = (krow >> 1) & 1;
    int nt = col0 >> 4;
    int ln = 16 * hb + (col0 & 15);
    int dst = ((kb * 8 + nt) * 32 + ln) * 2 + c;
    Wl[dst + 0] = w.x;
    Wl[dst + 2] = w.y;
    Wl[dst + 4] = w.z;
    Wl[dst + 6] = w.w;
  }

  // ---- RoPE-stage this wave's 16 rows via the (cos,sin) table ------------
  const int tile = blockIdx.x * 8 + wave;
  const long row0 = (long)tile * 16;
  const bool is_k = (row0 >= (long)ROWS_PER_TENSOR);
  const float* __restrict__ src = is_k ? k : q;
  const long srow0 = is_k ? (row0 - ROWS_PER_TENSOR) : row0;

  float* at = At[wave];
#pragma unroll 4
  for (int r = 0; r < 16; ++r) {
    long grow = srow0 + r;
    int n = (int)(grow & (NPOS - 1));
    v4f xy = *(const v4f*)(src + grow * DM + 4 * lane);       // 2 pairs
    v4f sc = *(const v4f*)(tab + n * (2 * 64) + 4 * lane);    // {c0,s0,c1,s1}
    v4f rot;
    rot.x = xy.x * sc.x - xy.y * sc.y;
    rot.y = xy.x * sc.y + xy.y * sc.x;
    rot.z = xy.z * sc.z - xy.w * sc.w;
    rot.w = xy.z * sc.w + xy.w * sc.z;
    *(v4f*)(at + r * AT_STRIDE + 4 * lane) = rot;
  }
  __syncthreads();  // Wl visible to all waves; At ordering within wave

  // ---- 16x128 @ 128x128 via v_wmma_f32_16x16x4_f32 -----------------------
  v8f acc[8] = {};
  const int m = lane & 15;
  const int hi = lane >> 4;

#pragma unroll 4
  for (int kb = 0; kb < 32; ++kb) {
    // A frag: lane holds (M = lane&15, K = 4*kb + 2*(lane>>4) + {0,1})
    v2f a = *(const v2f*)(at + m * AT_STRIDE + 4 * kb + 2 * hi);
#pragma unroll
    for (int nt = 0; nt < 8; ++nt) {
      v2f b = *(const v2f*)(Wl + (((kb * 8 + nt) * 32 + lane) << 1));
      acc[nt] = __builtin_amdgcn_wmma_f32_16x16x4_f32(
          /*neg_a=*/false, a, /*neg_b=*/false, b,
          /*c_mod=*/(short)0, acc[nt], /*reuse_a=*/false, /*reuse_b=*/false);
    }
  }

  // ---- Store: C/D layout -> element (M = v + 8*(lane>>4), N = lane&15) ---
#pragma unroll
  for (int nt = 0; nt < 8; ++nt) {
    int col = 16 * nt + m;
    long rbase = row0 + 8 * hi;
#pragma unroll
    for (int v = 0; v < 8; ++v) {
      out[(rbase + v) * DM + col] = acc[nt][v];
    }
  }
}

// ---------------------------------------------------------------------------
extern "C" void kernel_launch(void* const* d_in, const int* in_sizes, int n_in,
                              void* d_out, int out_size, void* d_ws,
                              size_t ws_size, hipStream_t stream) {
  const float* q = (const float*)d_in[0];
  const float* k = (const float*)d_in[1];
  const float* pos = (const float*)d_in[2];
  const float* freqs = (const float*)d_in[3];
  const float* sp = (const float*)d_in[4];
  float* out = (float*)d_out;
  float* W = (float*)d_ws;            // 128*128 f32 = 64 KB
  float* tab = W + DM * DM;           // 8192*64*2 f32 = 4 MB (cos,sin) table

  hipLaunchKernelGGL(cayley_prep_kernel, dim3(1 + TAB_BLOCKS), dim3(256), 0,
                     stream, sp, pos, freqs, W, tab);
  hipLaunchKernelGGL(rope_cayley_gemm_kernel, dim3(TILES / 8), dim3(256), 0,
                     stream, q, k, tab, W, out);
}